// Fusion_loss_17789754540341
// MI455X (gfx1250) — compile-verified
//
#include <hip/hip_runtime.h>

#define IMG_H 768
#define IMG_W 768
#define BATCH 16
#define TS 32          // output tile
#define HTS 34         // tile + 1px halo each side
#define LSTRIDE 34     // packed LDS row stride (floats) -- matches TDM packed writes
#define NPIX ((double)BATCH * IMG_H * IMG_W)

typedef float v2f __attribute__((ext_vector_type(2)));
typedef float v8f __attribute__((ext_vector_type(8)));
typedef unsigned int u32x4 __attribute__((ext_vector_type(4)));
typedef unsigned int u32x8 __attribute__((ext_vector_type(8)));

#if __has_builtin(__builtin_amdgcn_global_load_async_to_lds_b32)
#define HAVE_ASYNC_LDS 1
#else
#define HAVE_ASYNC_LDS 0
#endif

typedef __attribute__((address_space(1))) int* gptr_i32;
typedef __attribute__((address_space(3))) int* lptr_i32;

__device__ __forceinline__ void wait_async_all() {
#if __has_builtin(__builtin_amdgcn_s_wait_asynccnt)
  __builtin_amdgcn_s_wait_asynccnt(0);
#else
  asm volatile("s_wait_asynccnt 0x0" ::: "memory");
#endif
}

__device__ __forceinline__ void wait_tensor_all() {
#if __has_builtin(__builtin_amdgcn_s_wait_tensorcnt)
  __builtin_amdgcn_s_wait_tensorcnt(0);
#else
  asm volatile("s_wait_tensorcnt 0x0" ::: "memory");
#endif
}

// One 2-D TDM descriptor: copy tile_h rows of tile_w f32 elements (row stride
// row_stride_elems in global memory) into packed LDS at lds_byte_off.
// D# group0/group1 packed per CDNA5 ISA 8.3/8.4; groups 2/3 omitted (<=2D form).
__device__ __forceinline__ void tdm_load_2d(const float* gsrc,
                                            unsigned lds_byte_off,
                                            unsigned tensor_w_rem,
                                            unsigned tensor_h_rem,
                                            unsigned tile_w, unsigned tile_h,
                                            unsigned row_stride_elems) {
  const unsigned long long ga = (unsigned long long)(uintptr_t)gsrc;
  u32x4 g0;
  g0[0] = 1u;                                            // count=1 (valid), user mode
  g0[1] = lds_byte_off;                                  // lds_addr (bytes)
  g0[2] = (unsigned)(ga & 0xFFFFFFFFu);                  // global_addr[31:0]
  g0[3] = ((unsigned)(ga >> 32) & 0x01FFFFFFu) | (2u << 30);  // addr[56:32] | type=2
  u32x8 g1;
  g1[0] = (2u << 16);                                    // data_size=2 (4B); no mask/pad
  g1[1] = (tensor_w_rem & 0xFFFFu) << 16;                // tensor_dim0[15:0]
  g1[2] = ((tensor_w_rem >> 16) & 0xFFFFu) |             // tensor_dim0[31:16]
          ((tensor_h_rem & 0xFFFFu) << 16);              // tensor_dim1[15:0]
  g1[3] = ((tensor_h_rem >> 16) & 0xFFFFu) |             // tensor_dim1[31:16]
          (tile_w << 16);                                // tile_dim0
  g1[4] = tile_h;                                        // tile_dim1 (tile_dim2=0)
  g1[5] = row_stride_elems;                              // tensor_dim0_stride[31:0]
  g1[6] = 0;                                             // stride0[47:32], stride1[15:0]
  g1[7] = 0;
  asm volatile("tensor_load_to_lds %0, %1" ::"s"(g0), "s"(g1) : "memory");
}

// Wave32 sum reduction on the matrix pipe:
// A(16x4) per-lane = (v, 0)  -> row m = v(lane m) + v(lane m+16)
// B(4x16) = ones             -> D rows are cross-lane partial sums
__device__ __forceinline__ float wave_reduce_sum(float v) {
#if __has_builtin(__builtin_amdgcn_wmma_f32_16x16x4_f32)
  v2f a; a[0] = v;   a[1] = 0.0f;
  v2f b; b[0] = 1.0f; b[1] = 1.0f;
  v8f c = {0.f, 0.f, 0.f, 0.f, 0.f, 0.f, 0.f, 0.f};
  c = __builtin_amdgcn_wmma_f32_16x16x4_f32(false, a, false, b, (short)0, c,
                                            false, false);
  float s = ((c[0] + c[1]) + (c[2] + c[3])) + ((c[4] + c[5]) + (c[6] + c[7]));
  s += __shfl_xor(s, 16, 32);
  return s;
#else
  for (int off = 16; off > 0; off >>= 1) v += __shfl_xor(v, off, 32);
  return v;
#endif
}

__global__ __launch_bounds__(256) void sobel_loss_kernel(
    const float* __restrict__ vis, const float* __restrict__ ir,
    const float* __restrict__ X, const float* __restrict__ Y,
    double* __restrict__ acc) {
  __shared__ float lds[4][HTS * LSTRIDE];
  __shared__ float wpart[3][8];

  const int tid = threadIdx.x;
  const int tile_x = blockIdx.x, tile_y = blockIdx.y, bat = blockIdx.z;
  const int x0 = tile_x * TS - 1;
  const int y0 = tile_y * TS - 1;
  const float* imgs[4] = {vis, ir, X, Y};
  const size_t base = (size_t)bat * (IMG_H * IMG_W);

  const bool border = (tile_x == 0) | (tile_y == 0) |
                      (tile_x == (int)gridDim.x - 1) |
                      (tile_y == (int)gridDim.y - 1);

  if (!border) {
    // Interior tile: 4 TDM descriptors (one per image), issued by wave 0 only.
    // TDM ignores EXEC and is a wave-level op; TENSORcnt tracks completion.
    if (tid < 32) {
      const unsigned wrem = (unsigned)(IMG_W - x0);  // >= 65 > tile_w: no OOB
      const unsigned hrem = (unsigned)(IMG_H - y0);
#pragma unroll
      for (int im = 0; im < 4; ++im) {
        const float* src = imgs[im] + base + (size_t)y0 * IMG_W + x0;
        const unsigned loff = (unsigned)(uintptr_t)(lptr_i32)(&lds[im][0]);
        tdm_load_2d(src, loff, wrem, hrem, HTS, HTS, IMG_W);
      }
      wait_tensor_all();
    }
    __syncthreads();
  } else {
    // Border tile: zero-fill (SAME/zero padding) + per-lane async VMEM->LDS.
    for (int i = tid; i < HTS * LSTRIDE; i += 256) {
      lds[0][i] = 0.f; lds[1][i] = 0.f; lds[2][i] = 0.f; lds[3][i] = 0.f;
    }
    __syncthreads();
    for (int i = tid; i < HTS * HTS; i += 256) {
      const int ly = i / HTS, lx = i - ly * HTS;
      const int gy = y0 + ly, gx = x0 + lx;
      if ((unsigned)gy < (unsigned)IMG_H && (unsigned)gx < (unsigned)IMG_W) {
        const size_t goff = base + (size_t)gy * IMG_W + gx;
        const int loff = ly * LSTRIDE + lx;
#if HAVE_ASYNC_LDS
#pragma unroll
        for (int im = 0; im < 4; ++im) {
          __builtin_amdgcn_global_load_async_to_lds_b32(
              (gptr_i32)(imgs[im] + goff), (lptr_i32)(&lds[im][loff]), 0, 0);
        }
#else
#pragma unroll
        for (int im = 0; im < 4; ++im) lds[im][loff] = imgs[im][goff];
#endif
      }
    }
#if HAVE_ASYNC_LDS
    wait_async_all();
#endif
    __syncthreads();
  }

  // Each thread: 4 consecutive rows of one column. Horizontal partials per row
  // (rA = hr-hl, rB = hl+2hc+hr) are shared by vertically adjacent pixels:
  //   sx(i) = rA[i] + 2*rA[i+1] + rA[i+2],  sy(i) = rB[i] - rB[i+2].
  const int tx = tid & 31;   // column within tile
  const int ty = tid >> 5;   // row-group 0..7
  const int lx = tx + 1;
  const int py0 = ty * 4;    // first output row of this thread
  float s[4][4];
#pragma unroll
  for (int im = 0; im < 4; ++im) {
    const float* t = lds[im];
    float rA[6], rB[6];
#pragma unroll
    for (int j = 0; j < 6; ++j) {
      const float* row = t + (py0 + j) * LSTRIDE + lx;  // LDS rows py0 .. py0+5
      const float hl = row[-1], hc = row[0], hr = row[1];
      rA[j] = hr - hl;
      rB[j] = (hl + hr) + 2.f * hc;
    }
#pragma unroll
    for (int i = 0; i < 4; ++i) {
      const float sx = rA[i] + 2.f * rA[i + 1] + rA[i + 2];
      const float sy = rB[i] - rB[i + 2];
      s[im][i] = fabsf(sx) + fabsf(sy);
    }
  }
  float ax2 = 0.f, ay2 = 0.f, aab = 0.f;
#pragma unroll
  for (int i = 0; i < 4; ++i) {
    ax2 += s[2][i] * s[2][i];                         // sobel(X)^2
    ay2 += s[3][i] * s[3][i];                         // sobel(Y)^2
    aab += fabsf(s[2][i] - fmaxf(s[0][i], s[1][i]));  // |sobel(X)-max(vis,ir)|
  }

  // Wave-level reduction on the WMMA pipe (EXEC all-ones: uniform flow here).
  const float rx = wave_reduce_sum(ax2);
  const float ry = wave_reduce_sum(ay2);
  const float ra = wave_reduce_sum(aab);
  const int wave = tid >> 5;
  if ((tid & 31) == 0) {
    wpart[0][wave] = rx; wpart[1][wave] = ry; wpart[2][wave] = ra;
  }
  __syncthreads();
  if (tid == 0) {
    float sx = 0.f, sy = 0.f, sa = 0.f;
#pragma unroll
    for (int wv = 0; wv < 8; ++wv) {
      sx += wpart[0][wv]; sy += wpart[1][wv]; sa += wpart[2][wv];
    }
    atomicAdd(&acc[0], (double)sx);  // sum sobel(X)^2
    atomicAdd(&acc[1], (double)sy);  // sum sobel(Y)^2
    atomicAdd(&acc[2], (double)sa);  // sum |...|
  }
}

__global__ void init_acc_kernel(double* acc) {
  if (threadIdx.x < 3) acc[threadIdx.x] = 0.0;
}

__global__ void finalize_kernel(const double* __restrict__ acc,
                                float* __restrict__ out) {
  if (threadIdx.x == 0) {
    out[0] = (float)(acc[1] / acc[0]);  // loss_in = mean(Y^2)/mean(X^2)
    out[1] = (float)(acc[2] / NPIX);    // loss_grad = mean L1
  }
}

extern "C" void kernel_launch(void* const* d_in, const int* in_sizes, int n_in,
                              void* d_out, int out_size, void* d_ws,
                              size_t ws_size, hipStream_t stream) {
  const float* vis = (const float*)d_in[0];
  const float* ir  = (const float*)d_in[1];
  const float* X   = (const float*)d_in[2];
  const float* Y   = (const float*)d_in[3];
  double* acc = (double*)d_ws;
  float* out = (float*)d_out;

  hipLaunchKernelGGL(init_acc_kernel, dim3(1), dim3(32), 0, stream, acc);
  dim3 grid(IMG_W / TS, IMG_H / TS, BATCH);  // 24 x 24 x 16
  hipLaunchKernelGGL(sobel_loss_kernel, grid, dim3(256), 0, stream, vis, ir, X,
                     Y, acc);
  hipLaunchKernelGGL(finalize_kernel, dim3(1), dim3(32), 0, stream, acc, out);
}